// GraphMemoryModule_9964324127104
// MI455X (gfx1250) — compile-verified
//
#include <hip/hip_runtime.h>
#include <math.h>
#include <stdint.h>

#define N_NODES  100000
#define N_EDGES  1600000
#define NODE_DIM 128
#define MEMD     64
#define NEG_SLOPE 0.2f
#define E_TOT    (N_EDGES + N_NODES)
#define N64      (N_NODES * MEMD)

typedef __attribute__((ext_vector_type(2))) float v2f;
typedef __attribute__((ext_vector_type(8))) float v8f;

// ---------- monotonic float<->uint encoding for atomicMax on floats ----------
__device__ __forceinline__ unsigned fenc(float f) {
  unsigned u = __float_as_uint(f);
  return (u & 0x80000000u) ? ~u : (u | 0x80000000u);
}
__device__ __forceinline__ float fdec(unsigned u) {
  return (u & 0x80000000u) ? __uint_as_float(u & 0x7FFFFFFFu) : __uint_as_float(~u);
}

// ---------- CDNA5 async global->LDS copy (ASYNCcnt path) ----------
// VDST = per-lane LDS byte offset, VADDR = per-lane 64-bit global address.
__device__ __forceinline__ void async_copy_b128(const float* g, float* l) {
  unsigned loff = (unsigned)(size_t)l;        // low 32 bits of generic = LDS offset
  uint64_t ga = (uint64_t)(size_t)g;
  asm volatile("global_load_async_to_lds_b128 %0, %1, off"
               :: "v"(loff), "v"(ga) : "memory");
}
__device__ __forceinline__ void async_wait0() {
  asm volatile("s_wait_asynccnt 0x0" ::: "memory");
}

// ---------- one 16x16 f32 WMMA tile:  A(LDS)[16,K] @ W(global)[N,K]^T ----------
// A layout (16x4 f32/step): lanes 0-15 -> M=lane, K=k..k+1; lanes 16-31 -> K=k+2..k+3.
template <int K, int LDA>
__device__ __forceinline__ v8f wmma_tile_lds(const float* sA,
                                             const float* __restrict__ W, int col0) {
  const int lane = threadIdx.x & 31;
  const int mi   = lane & 15;
  const int kh   = (lane >> 4) << 1;
  const float* ap = sA + mi * LDA + kh;
  const float* wp = W + (size_t)(col0 + mi) * K + kh;
  v8f c = {0.f, 0.f, 0.f, 0.f, 0.f, 0.f, 0.f, 0.f};
#pragma unroll
  for (int k = 0; k < K; k += 4) {
    v2f a = *(const v2f*)(ap + k);
    v2f b = *(const v2f*)(wp + k);
    c = __builtin_amdgcn_wmma_f32_16x16x4_f32(false, a, false, b,
                                              (short)0, c, false, false);
  }
  return c;
}

// ---------- kernel 0: init scratch (ctx=0, max=-inf, denom=0) ----------
__global__ void k_init(float* ctx, unsigned* mx, float* denom) {
  int i = blockIdx.x * blockDim.x + threadIdx.x;
  if (i < N64) ctx[i] = 0.f;
  if (i < N_NODES) { mx[i] = fenc(-__builtin_huge_valf()); denom[i] = 0.f; }
}

// ---------- kernel 1 (fused front-end):
//   mem_enh = state @ w_in^T + b_in + goal + mem     (WMMA, K=128, A via async->LDS)
//   xt      = mem_enh @ w_gat^T                      (WMMA, K=64, A from LDS)
//   a_s     = xt @ att_src ; a_d = xt @ att_dst      (LDS dots)
// block = 128 threads (4 waves) handling 16 rows.
__global__ void k_front(const float* __restrict__ state, const float* __restrict__ goal,
                        const float* __restrict__ mem,   const float* __restrict__ w_in,
                        const float* __restrict__ b_in,  const float* __restrict__ w_gat,
                        const float* __restrict__ att_s, const float* __restrict__ att_d,
                        float* __restrict__ mem_enh, float* __restrict__ xt,
                        float* __restrict__ a_s, float* __restrict__ a_d) {
  __shared__ float sA[16 * 132];   // state tile 16x128 (pad->conflict-free frag reads)
  __shared__ float sE[16 * 68];    // mem_enh tile 16x64
  __shared__ float sX[16 * 68];    // xt tile 16x64
  const int row0 = blockIdx.x * 16;
  const int t = threadIdx.x;

  // stage the 16x128 A tile via async DMA (4 passes x 128 lanes x 16B)
#pragma unroll
  for (int pass = 0; pass < 4; ++pass) {
    int r = pass * 4 + (t >> 5);
    int c = (t & 31) * 4;
    async_copy_b128(state + (size_t)(row0 + r) * NODE_DIM + c, sA + r * 132 + c);
  }
  __builtin_prefetch(w_gat + t * 32, 0, 0);   // warm stage-2 weights (global_prefetch_b8)
  async_wait0();
  __syncthreads();

  const int wave = t >> 5, lane = t & 31;
  const int mi = lane & 15, mo = (lane >> 4) * 8;
  const int col0 = wave * 16;

  // stage 1: projection tile
  v8f c1 = wmma_tile_lds<NODE_DIM, 132>(sA, w_in, col0);
#pragma unroll
  for (int v = 0; v < 8; ++v) {
    int cc = col0 + mi;
    size_t o = (size_t)(row0 + mo + v) * MEMD + cc;
    float val = c1[v] + b_in[cc] + goal[o] + mem[o];
    mem_enh[o] = val;
    sE[(mo + v) * 68 + cc] = val;
  }
  __syncthreads();

  // stage 2: GAT linear tile straight from LDS
  v8f c2 = wmma_tile_lds<MEMD, 68>(sE, w_gat, col0);
#pragma unroll
  for (int v = 0; v < 8; ++v) {
    int cc = col0 + mi;
    xt[(size_t)(row0 + mo + v) * MEMD + cc] = c2[v];
    sX[(mo + v) * 68 + cc] = c2[v];
  }
  __syncthreads();

  // stage 3: attention coefficients (row dots from LDS)
  if (t < 32) {
    int r = t & 15;
    const float* xr = sX + r * 68;
    const float* av = (t < 16) ? att_s : att_d;
    float acc = 0.f;
#pragma unroll
    for (int c = 0; c < MEMD; ++c) acc += xr[c] * av[c];
    if (t < 16) a_s[row0 + r] = acc;
    else        a_d[row0 + r] = acc;
  }
}

// ---------- kernel 2: e = leaky(a_s[src]+a_d[dst]); segment max over dst ----------
__global__ void k_edge_max(const int* __restrict__ ei, const float* __restrict__ a_s,
                           const float* __restrict__ a_d, unsigned* __restrict__ mx,
                           float* __restrict__ e_buf) {
  int idx = blockIdx.x * blockDim.x + threadIdx.x;
  if (idx >= E_TOT) return;
  int src, dst;
  if (idx < N_EDGES) { src = ei[idx]; dst = ei[N_EDGES + idx]; }
  else               { src = dst = idx - N_EDGES; }          // self loops
  float e = a_s[src] + a_d[dst];
  e = (e >= 0.f) ? e : NEG_SLOPE * e;
  e_buf[idx] = e;
  atomicMax(mx + dst, fenc(e));
}

// ---------- kernel 3: ex = exp(e - m[dst]); denom[dst] += ex ----------
__global__ void k_edge_exp(const int* __restrict__ ei, const unsigned* __restrict__ mx,
                           float* __restrict__ e_buf, float* __restrict__ denom) {
  int idx = blockIdx.x * blockDim.x + threadIdx.x;
  if (idx >= E_TOT) return;
  int dst = (idx < N_EDGES) ? ei[N_EDGES + idx] : (idx - N_EDGES);
  float v = __expf(e_buf[idx] - fdec(mx[dst]));
  e_buf[idx] = v;
  atomicAdd(denom + dst, v);
}

// ---------- kernel 4: ctx[dst] += alpha * xt[src]  (wave per edge) ----------
__global__ void k_edge_scatter(const int* __restrict__ ei, const float* __restrict__ e_buf,
                               const float* __restrict__ denom, const float* __restrict__ xt,
                               float* __restrict__ ctx) {
  int edge = (int)((blockIdx.x * (size_t)blockDim.x + threadIdx.x) >> 5);
  int lane = threadIdx.x & 31;
  if (edge >= E_TOT) return;
  int src, dst;
  if (edge < N_EDGES) { src = ei[edge]; dst = ei[N_EDGES + edge]; }
  else                { src = dst = edge - N_EDGES; }
  float alpha = e_buf[edge] / denom[dst];
  const float* xr = xt + (size_t)src * MEMD;
  float* cr = ctx + (size_t)dst * MEMD;
  atomicAdd(cr + lane,      alpha * xr[lane]);
  atomicAdd(cr + lane + 32, alpha * xr[lane + 32]);
}

// ---------- kernel 5: memory_input = mem_enh + ctx + b_gat (in place) ----------
__global__ void k_fin(float* __restrict__ mem_enh, const float* __restrict__ ctx,
                      const float* __restrict__ b_gat) {
  int i = blockIdx.x * blockDim.x + threadIdx.x;
  if (i >= N64) return;
  mem_enh[i] = mem_enh[i] + ctx[i] + b_gat[i & (MEMD - 1)];
}

// ---------- kernel 6: fused GRU cell ----------
// x,h tiles async-staged to LDS; WMMA gi/gh -> LDS -> gates; h reused from LDS.
#define GST 204
__global__ void k_gru(const float* __restrict__ x, const float* __restrict__ h,
                      const float* __restrict__ w_ih, const float* __restrict__ w_hh,
                      const float* __restrict__ b_ih, const float* __restrict__ b_hh,
                      float* __restrict__ out) {
  __shared__ float sx[16 * 68];
  __shared__ float sh[16 * 68];
  __shared__ float gi[16 * GST];
  __shared__ float gh[16 * GST];
  const int row0 = blockIdx.x * 16;
  const int t = threadIdx.x;

#pragma unroll
  for (int pass = 0; pass < 2; ++pass) {
    int r = pass * 8 + (t >> 4);
    int c = (t & 15) * 4;
    async_copy_b128(x + (size_t)(row0 + r) * MEMD + c, sx + r * 68 + c);
    async_copy_b128(h + (size_t)(row0 + r) * MEMD + c, sh + r * 68 + c);
  }
  async_wait0();
  __syncthreads();

  const int wave = t >> 5, lane = t & 31;
  const int mi = lane & 15, mo = (lane >> 4) * 8;
#pragma unroll
  for (int ti = 0; ti < 3; ++ti) {
    int col0 = wave * 48 + ti * 16;
    v8f ci = wmma_tile_lds<MEMD, 68>(sx, w_ih, col0);
#pragma unroll
    for (int v = 0; v < 8; ++v) gi[(mo + v) * GST + col0 + mi] = ci[v];
    v8f ch = wmma_tile_lds<MEMD, 68>(sh, w_hh, col0);
#pragma unroll
    for (int v = 0; v < 8; ++v) gh[(mo + v) * GST + col0 + mi] = ch[v];
  }
  __syncthreads();

#pragma unroll
  for (int j = 0; j < 8; ++j) {
    int p = t * 8 + j;                        // 0..1023
    int row = p >> 6, col = p & 63;
    const float* gir = gi + row * GST;
    const float* ghr = gh + row * GST;
    float ir = gir[col]        + b_ih[col];
    float iz = gir[64 + col]   + b_ih[64 + col];
    float in = gir[128 + col]  + b_ih[128 + col];
    float hr = ghr[col]        + b_hh[col];
    float hz = ghr[64 + col]   + b_hh[64 + col];
    float hn = ghr[128 + col]  + b_hh[128 + col];
    float r = 1.f / (1.f + __expf(-(ir + hr)));
    float z = 1.f / (1.f + __expf(-(iz + hz)));
    float n = tanhf(in + r * hn);
    float hv = sh[row * 68 + col];
    out[(size_t)(row0 + row) * MEMD + col] = (1.f - z) * n + z * hv;
  }
}

extern "C" void kernel_launch(void* const* d_in, const int* in_sizes, int n_in,
                              void* d_out, int out_size, void* d_ws, size_t ws_size,
                              hipStream_t stream) {
  const float* state   = (const float*)d_in[0];
  const float* goal    = (const float*)d_in[1];
  const float* mem     = (const float*)d_in[2];
  const int*   ei      = (const int*)d_in[3];    // int32 (JAX x64 disabled)
  const float* w_in    = (const float*)d_in[4];
  const float* b_in    = (const float*)d_in[5];
  const float* w_gat   = (const float*)d_in[6];
  const float* att_src = (const float*)d_in[7];
  const float* att_dst = (const float*)d_in[8];
  const float* b_gat   = (const float*)d_in[9];
  const float* w_ih    = (const float*)d_in[10];
  const float* w_hh    = (const float*)d_in[11];
  const float* b_ih    = (const float*)d_in[12];
  const float* b_hh    = (const float*)d_in[13];
  float* out = (float*)d_out;

  float* ws = (float*)d_ws;
  float*    mem_enh = ws;                         // [N,64] -> becomes memory_input
  float*    xt      = ws + (size_t)N64;           // [N,64]
  float*    a_s     = ws + 2 * (size_t)N64;       // [N]
  float*    a_d     = a_s + N_NODES;              // [N]
  unsigned* mx      = (unsigned*)(a_d + N_NODES); // [N]
  float*    denom   = (float*)mx + N_NODES;       // [N]
  float*    ctx     = denom + N_NODES;            // [N,64]
  float*    e_buf   = ctx + (size_t)N64;          // [E+N]

  const int gMM   = N_NODES / 16;                 // 6250 (exact)
  const int gInit = (N64 + 255) / 256;
  const int gEdge = (E_TOT + 255) / 256;
  const int gScat = (E_TOT + 7) / 8;

  k_init<<<gInit, 256, 0, stream>>>(ctx, mx, denom);
  k_front<<<gMM, 128, 0, stream>>>(state, goal, mem, w_in, b_in, w_gat,
                                   att_src, att_dst, mem_enh, xt, a_s, a_d);
  k_edge_max<<<gEdge, 256, 0, stream>>>(ei, a_s, a_d, mx, e_buf);
  k_edge_exp<<<gEdge, 256, 0, stream>>>(ei, mx, e_buf, denom);
  k_edge_scatter<<<gScat, 256, 0, stream>>>(ei, e_buf, denom, xt, ctx);
  k_fin<<<gInit, 256, 0, stream>>>(mem_enh, ctx, b_gat);
  k_gru<<<gMM, 128, 0, stream>>>(mem_enh, mem, w_ih, w_hh, b_ih, b_hh, out);
}